// GNO_73830487818865
// MI455X (gfx1250) — compile-verified
//
#include <hip/hip_runtime.h>
#include <math.h>

// ---------------------------------------------------------------------------
// Galerkin-type neural operator, restructured via associativity:
//   (q k^T) v  ==  q (k^T v)   -> O(N d^2) instead of O(N^2 d)
// All GEMMs done in fp32 with V_WMMA_F32_16X16X4_F32 (wave32, CDNA5).
// ---------------------------------------------------------------------------

#define HID    64
#define NPB    4096            // tokens per batch (64*64)
#define BATCH  2
#define MROWS  (BATCH * NPB)   // 8192 total rows
#define KCH    256             // N-chunk per wave in k^T v reduction

typedef __attribute__((ext_vector_type(2))) float v2f;
typedef __attribute__((ext_vector_type(8))) float v8f;

__device__ __forceinline__ v8f wmma4(v2f a, v2f b, v8f c) {
  // D = A(16x4,f32) * B(4x16,f32) + C(16x16,f32)
  return __builtin_amdgcn_wmma_f32_16x16x4_f32(
      /*neg_a=*/false, a, /*neg_b=*/false, b,
      /*c_mod=*/(short)0, c, /*reuse_a=*/false, /*reuse_b=*/false);
}

__device__ __forceinline__ v8f vzero8() {
  v8f z = {0.f, 0.f, 0.f, 0.f, 0.f, 0.f, 0.f, 0.f};
  return z;
}

// --------------------------- lift: x[M,3] -> h[M,64] ------------------------
__global__ void lift_kernel(const float* __restrict__ x,
                            const float* __restrict__ lw,
                            const float* __restrict__ lb,
                            float* __restrict__ h) {
  int idx = blockIdx.x * blockDim.x + threadIdx.x;   // M*64 threads
  int m = idx >> 6;
  int c = idx & 63;
  float s = lb[c]
          + x[m * 3 + 0] * lw[0 * HID + c]
          + x[m * 3 + 1] * lw[1 * HID + c]
          + x[m * 3 + 2] * lw[2 * HID + c];
  h[idx] = s;
}

// ------------- fused q/k/v/wh projections: h[M,64] @ W[64,64] + b -----------
// One wave per 16x16 output tile of the logical [M, 4*64] output.
__global__ void proj4_kernel(const float* __restrict__ h,
                             const float* __restrict__ wq, const float* __restrict__ bq,
                             const float* __restrict__ wk, const float* __restrict__ bk,
                             const float* __restrict__ wv, const float* __restrict__ bv,
                             const float* __restrict__ wb, const float* __restrict__ bb,
                             float* __restrict__ outbase) {
  int wave = (blockIdx.x * blockDim.x + threadIdx.x) >> 5;
  int lane = threadIdx.x & 31;
  int li = lane & 15;
  int hi = lane >> 4;

  int mt = wave >> 4;               // 0..(M/16 - 1)
  int nt = wave & 15;               // 0..15 across the 4*64 output columns
  int which = nt >> 2;              // 0=q 1=k 2=v 3=wh
  int col = ((nt & 3) << 4) + li;   // 0..63

  const float* W  = (which == 0) ? wq : (which == 1) ? wk : (which == 2) ? wv : wb;
  const float* Bp = (which == 0) ? bq : (which == 1) ? bk : (which == 2) ? bv : bb;

  int m0 = mt << 4;
  const float* arow = h + (size_t)(m0 + li) * HID;

  v8f acc = vzero8();
#pragma unroll
  for (int k0 = 0; k0 < HID; k0 += 4) {
    int ka = k0 + (hi << 1);
    v2f a = { arow[ka], arow[ka + 1] };
    v2f b = { W[ka * HID + col], W[(ka + 1) * HID + col] };
    acc = wmma4(a, b, acc);
  }

  float bias = Bp[col];
  float* outp = outbase + (size_t)which * MROWS * HID;
#pragma unroll
  for (int j = 0; j < 8; ++j) {
    int row = m0 + (hi << 3) + j;
    outp[(size_t)row * HID + col] = acc[j] + bias;
  }
}

// -------------------- S[b] = k_b^T @ v_b  (64x64 per batch) -----------------
// K dimension = tokens; chunked over N, accumulated with f32 global atomics.
__global__ void ktv_kernel(const float* __restrict__ kbuf,
                           const float* __restrict__ vbuf,
                           float* __restrict__ S) {
  int wave = (blockIdx.x * blockDim.x + threadIdx.x) >> 5;
  int lane = threadIdx.x & 31;
  int li = lane & 15;
  int hi = lane >> 4;

  int tile  = wave & 15;            // 4x4 tiles covering 64x64
  int chunk = (wave >> 4) & 15;     // NPB / KCH = 16 chunks
  int b     = wave >> 8;            // batch
  int it = tile >> 2;
  int jt = tile & 3;

  const float* kp = kbuf + (size_t)b * NPB * HID;
  const float* vp = vbuf + (size_t)b * NPB * HID;
  int i = (it << 4) + li;           // output row  (A-frag lane index = M)
  int j = (jt << 4) + li;           // output col  (B-frag lane index = N)

  v8f acc = vzero8();
  int n0 = chunk * KCH;
  for (int n = n0; n < n0 + KCH; n += 4) {
    int na = n + (hi << 1);
    v2f a  = { kp[(size_t)na * HID + i], kp[(size_t)(na + 1) * HID + i] };
    v2f bb = { vp[(size_t)na * HID + j], vp[(size_t)(na + 1) * HID + j] };
    acc = wmma4(a, bb, acc);
  }

  float* sp = S + (size_t)b * HID * HID;
#pragma unroll
  for (int jj = 0; jj < 8; ++jj) {
    int row = (it << 4) + (hi << 3) + jj;
    atomicAdd(&sp[row * HID + (jt << 4) + li], acc[jj]);
  }
}

// -------- h = gelu( wh + (q @ S[b]) * scale / N ),  exact erf GELU ----------
__global__ void attn_gelu_kernel(const float* __restrict__ q,
                                 const float* __restrict__ S,
                                 const float* __restrict__ wh,
                                 float* __restrict__ hout) {
  int wave = (blockIdx.x * blockDim.x + threadIdx.x) >> 5;
  int lane = threadIdx.x & 31;
  int li = lane & 15;
  int hi = lane >> 4;

  int mt = wave >> 2;               // 0..(M/16 - 1)
  int nt = wave & 3;                // 64 cols / 16
  int m0 = mt << 4;
  int b  = m0 >> 12;                // m0 / 4096
  int col = (nt << 4) + li;

  const float* Sp   = S + (size_t)b * HID * HID;
  const float* arow = q + (size_t)(m0 + li) * HID;

  v8f acc = vzero8();
#pragma unroll
  for (int k0 = 0; k0 < HID; k0 += 4) {
    int ka = k0 + (hi << 1);
    v2f a  = { arow[ka], arow[ka + 1] };
    v2f bb = { Sp[ka * HID + col], Sp[(ka + 1) * HID + col] };
    acc = wmma4(a, bb, acc);
  }

  const float FACT = 3.0517578125e-05f;          // (1/sqrt(64)) / 4096
  const float INV_SQRT2 = 0.70710678118654752f;
#pragma unroll
  for (int jj = 0; jj < 8; ++jj) {
    int row = m0 + (hi << 3) + jj;
    float xv = wh[(size_t)row * HID + col] + acc[jj] * FACT;
    hout[(size_t)row * HID + col] = 0.5f * xv * (1.0f + erff(xv * INV_SQRT2));
  }
}

// ---------------------- out[m] = h[m,:] . proj_w + proj_b -------------------
__global__ void proj_out_kernel(const float* __restrict__ h,
                                const float* __restrict__ pw,
                                const float* __restrict__ pb,
                                float* __restrict__ out) {
  int m = blockIdx.x * blockDim.x + threadIdx.x;
  const float4* hr = (const float4*)(h + (size_t)m * HID);
  const float4* wr = (const float4*)pw;
  float s = 0.f;
#pragma unroll
  for (int i = 0; i < 16; ++i) {
    float4 a = hr[i];
    float4 w = wr[i];
    s += a.x * w.x + a.y * w.y + a.z * w.z + a.w * w.w;
  }
  out[m] = s + pb[0];
}

// ---------------------------------------------------------------------------
extern "C" void kernel_launch(void* const* d_in, const int* in_sizes, int n_in,
                              void* d_out, int out_size, void* d_ws, size_t ws_size,
                              hipStream_t stream) {
  const float* x      = (const float*)d_in[0];
  const float* lift_w = (const float*)d_in[1];
  const float* lift_b = (const float*)d_in[2];
  const float* blk_w  = (const float*)d_in[3];
  const float* blk_b  = (const float*)d_in[4];
  const float* q_w    = (const float*)d_in[5];
  const float* q_b    = (const float*)d_in[6];
  const float* k_w    = (const float*)d_in[7];
  const float* k_b    = (const float*)d_in[8];
  const float* v_w    = (const float*)d_in[9];
  const float* v_b    = (const float*)d_in[10];
  const float* proj_w = (const float*)d_in[11];
  const float* proj_b = (const float*)d_in[12];
  float* out = (float*)d_out;

  // workspace layout: h | q | k | v | wh | S   (~10.03 MB + 32 KB)
  float* h  = (float*)d_ws;
  float* qb = h  + (size_t)MROWS * HID;
  float* kb = qb + (size_t)MROWS * HID;
  float* vb = kb + (size_t)MROWS * HID;
  float* wh = vb + (size_t)MROWS * HID;
  float* S  = wh + (size_t)MROWS * HID;

  // lift: M*64 threads
  lift_kernel<<<(MROWS * HID) / 256, 256, 0, stream>>>(x, lift_w, lift_b, h);

  for (int L = 0; L < 4; ++L) {
    const int woff = L * HID * HID;
    const int boff = L * HID;
    // 8192 tiles (waves), 4 waves per block
    proj4_kernel<<<(MROWS / 16) * 16 / 4, 128, 0, stream>>>(
        h,
        q_w + woff,  q_b + boff,
        k_w + woff,  k_b + boff,
        v_w + woff,  v_b + boff,
        blk_w + woff, blk_b + boff,
        qb);
    hipMemsetAsync(S, 0, (size_t)BATCH * HID * HID * sizeof(float), stream);
    // 2 batches * 16 tiles * 16 chunks = 512 waves
    ktv_kernel<<<512 / 4, 128, 0, stream>>>(kb, vb, S);
    // 512 row-tiles * 4 col-tiles = 2048 waves
    attn_gelu_kernel<<<2048 / 4, 128, 0, stream>>>(qb, S, wh, h);
  }

  proj_out_kernel<<<MROWS / 256, 256, 0, stream>>>(h, proj_w, proj_b, out);
}